// FractalEmbeddingLayer_54039278518588
// MI455X (gfx1250) — compile-verified
//
#include <hip/hip_runtime.h>
#include <hip/hip_bf16.h>

typedef __attribute__((ext_vector_type(2))) float v2f;
typedef __attribute__((ext_vector_type(8))) float v8f;
typedef __attribute__((ext_vector_type(4))) int   v4i;
typedef __attribute__((address_space(1))) v4i     g_v4i;   // global-AS int4
typedef __attribute__((address_space(3))) v4i     l_v4i;   // LDS-AS int4

#define PATCH 16
#define EMB   64
#define FEAT  256                // PATCH*PATCH
#define CH    3
#define IMG   512
#define NPX   32                 // patches per side
#define PPB   (NPX * NPX * CH)   // 3072 patch-rows per batch image
#define MT    64                 // patch-rows per workgroup
#define THREADS 128
#define SP_STRIDE 260            // padded 256 floats per patch row
#define B_STRIDE  66             // padded 64 float2 per K-pair row

#define LDS_FLOATS (MT * SP_STRIDE + (FEAT / 2) * B_STRIDE * 2)
#define LDS_BYTES  (LDS_FLOATS * sizeof(float))

#define HAS_ASYNC_LDS __has_builtin(__builtin_amdgcn_global_load_async_to_lds_b128)

// ---------------------------------------------------------------------------
// Fused: async patch staging -> (WMMA f32 GEMM embedding) + (fractal boxes)
// ---------------------------------------------------------------------------
__global__ void fractal_embed_kernel(const float* __restrict__ x,
                                     const float* __restrict__ W,
                                     const float* __restrict__ bias,
                                     float* __restrict__ emb,
                                     float* __restrict__ fract) {
    extern __shared__ float smem[];
    float* sP = smem;                                    // [MT][SP_STRIDE]
    v2f*   sB = (v2f*)(smem + MT * SP_STRIDE);           // [FEAT/2][B_STRIDE] K-pairs

    const int tid    = threadIdx.x;
    const int wgBase = blockIdx.x * MT;                  // first global patch-row

    // ---- stage W as K-pair-interleaved B tiles: sB[f/2][e] = (W[e][f], W[e][f+1]) ----
    for (int i = 0; i < (EMB * FEAT / 4) / THREADS; ++i) {     // 32 iters
        int lin4 = tid + i * THREADS;              // float4 index into W
        int e    = lin4 >> 6;                      // embedding row
        int fq   = lin4 & 63;                      // feature quad (f = fq*4)
        const float4 v = *(const float4*)(W + (size_t)lin4 * 4);
        v2f lo; lo.x = v.x; lo.y = v.y;
        v2f hi; hi.x = v.z; hi.y = v.w;
        sB[(fq * 2 + 0) * B_STRIDE + e] = lo;
        sB[(fq * 2 + 1) * B_STRIDE + e] = hi;
    }

    // ---- stage 64 patch-rows into LDS (async global->LDS, no VGPR round trip) ----
    for (int i = 0; i < (MT * FEAT / 4) / THREADS; ++i) {      // 32 iters
        int lin    = tid + i * THREADS;            // [0, 4096) float4s
        int mLocal = lin >> 6;                     // patch-row within WG
        int within = lin & 63;
        int r      = within >> 2;                  // pixel row in patch
        int q      = within & 3;                   // float4 column
        int m   = wgBase + mLocal;
        int b   = m / PPB;
        int rem = m - b * PPB;
        int p   = rem / CH;
        int c   = rem - p * CH;
        int py  = p >> 5;
        int px  = p & 31;
        const float* src = x + (((size_t)(b * CH + c) * IMG + py * PATCH + r) * IMG
                                + px * PATCH + q * 4);
        float* dst = &sP[mLocal * SP_STRIDE + r * PATCH + q * 4];
#if HAS_ASYNC_LDS
        __builtin_amdgcn_global_load_async_to_lds_b128((g_v4i*)src, (l_v4i*)dst, 0, 0);
#else
        const float4 v = *(const float4*)src;
        dst[0] = v.x; dst[1] = v.y; dst[2] = v.z; dst[3] = v.w;
#endif
    }
#if HAS_ASYNC_LDS
    asm volatile("s_wait_asynccnt 0" ::: "memory");
#endif
    __syncthreads();

    // ---- WMMA GEMM: each of 4 waves computes a 16x64 tile ----
    const int wave  = tid >> 5;
    const int lane  = tid & 31;
    const int half  = lane >> 4;          // 0: lanes 0-15, 1: lanes 16-31
    const int l16   = lane & 15;
    const int mBase = wave * 16;

    v8f acc0 = {}, acc1 = {}, acc2 = {}, acc3 = {};
    const v2f* aRow = (const v2f*)&sP[(mBase + l16) * SP_STRIDE];

    for (int k = 0; k < FEAT / 4; ++k) {
        // pair index p2 covers features {4k+2*half, 4k+2*half+1} for this half-wave
        const int p2 = k * 2 + half;
        const v2f a = aRow[p2];                    // single ds_load_b64
        const v2f* bRow = sB + p2 * B_STRIDE;
        const v2f b0 = bRow[l16 +  0];
        const v2f b1 = bRow[l16 + 16];
        const v2f b2 = bRow[l16 + 32];
        const v2f b3 = bRow[l16 + 48];
        acc0 = __builtin_amdgcn_wmma_f32_16x16x4_f32(false, a, false, b0, (short)0, acc0, false, false);
        acc1 = __builtin_amdgcn_wmma_f32_16x16x4_f32(false, a, false, b1, (short)0, acc1, false, false);
        acc2 = __builtin_amdgcn_wmma_f32_16x16x4_f32(false, a, false, b2, (short)0, acc2, false, false);
        acc3 = __builtin_amdgcn_wmma_f32_16x16x4_f32(false, a, false, b3, (short)0, acc3, false, false);
    }

    // ---- epilogue: bias + store. C layout: VGPR v -> M = v + 8*half, N = l16 ----
    const float bv0 = bias[l16 +  0];
    const float bv1 = bias[l16 + 16];
    const float bv2 = bias[l16 + 32];
    const float bv3 = bias[l16 + 48];
#pragma unroll
    for (int v = 0; v < 8; ++v) {
        const size_t row = (size_t)(wgBase + mBase + v + half * 8) * EMB;
        emb[row +  0 + l16] = acc0[v] + bv0;
        emb[row + 16 + l16] = acc1[v] + bv1;
        emb[row + 32 + l16] = acc2[v] + bv2;
        emb[row + 48 + l16] = acc3[v] + bv3;
    }

    // ---- fractal dimension: one thread per patch-row, from LDS ----
    if (tid < MT) {
        const float* pp = &sP[tid * SP_STRIDE];
        unsigned rowbits[PATCH];
#pragma unroll
        for (int r = 0; r < PATCH; ++r) {
            unsigned bits = 0;
#pragma unroll
            for (int c = 0; c < PATCH; ++c)
                bits |= (pp[r * PATCH + c] > 0.1f) ? (1u << c) : 0u;
            rowbits[r] = bits;
        }
        float ys[4];
        int si = 0;
        for (int s = 2; s <= 16; s <<= 1, ++si) {
            const int nb = PATCH / s;
            const unsigned gmask = (s == 16) ? 0xFFFFu : ((1u << s) - 1u);
            int count = 0;
            for (int by = 0; by < nb; ++by) {
                unsigned orr = 0;
                for (int rr = 0; rr < s; ++rr) orr |= rowbits[by * s + rr];
                for (int bx = 0; bx < nb; ++bx)
                    count += ((orr >> (bx * s)) & gmask) ? 1 : 0;
            }
            ys[si] = __logf((float)count);
        }
        // xs = {-1,-2,-3,-4}*ln2 ; closed-form least-squares slope:
        const float slope = (3.0f * ys[0] + ys[1] - ys[2] - 3.0f * ys[3])
                            * (1.0f / (10.0f * 0.69314718056f));
        fract[wgBase + tid] = slope;
    }
}

// ---------------------------------------------------------------------------
// Harmonics: closed-form eigenbasis of the Kronecker-sum grid Laplacian.
// ---------------------------------------------------------------------------
__global__ void harmonics_kernel(float* __restrict__ harm) {
    const int idx = blockIdx.x * blockDim.x + threadIdx.x;   // [0, 2560)
    if (idx >= FEAT * 10) return;
    const int i = idx / 10;          // grid point index (a*16 + b)
    const int h = idx - 10 * i;      // harmonic index
    const int a = i >> 4;
    const int b = i & 15;
    const int jl[10] = {1, 1, 2, 2, 1, 3, 2, 3, 1, 4};
    const int kl[10] = {1, 2, 1, 2, 3, 1, 3, 2, 4, 1};
    const float PI17 = 3.14159265358979f / 17.0f;
    const float w = (2.0f / 17.0f)
                  * __sinf((float)((a + 1) * jl[h]) * PI17)
                  * __sinf((float)((b + 1) * kl[h]) * PI17);
    harm[idx] = w;
}

// ---------------------------------------------------------------------------
extern "C" void kernel_launch(void* const* d_in, const int* in_sizes, int n_in,
                              void* d_out, int out_size, void* d_ws, size_t ws_size,
                              hipStream_t stream) {
    const float* x    = (const float*)d_in[0];   // [32,3,512,512]
    const float* W    = (const float*)d_in[1];   // [64,256]
    const float* bias = (const float*)d_in[2];   // [64]

    float* out   = (float*)d_out;
    float* emb   = out;                               // [32*3072*64]
    float* fract = out + (size_t)32 * PPB * EMB;      // [32*3072]
    float* harm  = fract + (size_t)32 * PPB;          // [256*10]

    const int mTiles = (32 * PPB) / MT;               // 1536 workgroups
    fractal_embed_kernel<<<mTiles, THREADS, LDS_BYTES, stream>>>(x, W, bias, emb, fract);
    harmonics_kernel<<<(FEAT * 10 + 255) / 256, 256, 0, stream>>>(harm);
}